// Seq2SeqHGNN_65652870087169
// MI455X (gfx1250) — compile-verified
//
#include <hip/hip_runtime.h>
#include <hip/hip_bf16.h>

// ---------------------------------------------------------------------------
// Seq2SeqHGNN on MI455X (gfx1250): fp32 WMMA GEMMs + atomic scatter phases.
// B-matrix staged in LDS in K-pair-major float2 layout so each WMMA B fragment
// is one aligned, bank-conflict-free ds_load_b64 (no register shuffles).
// ---------------------------------------------------------------------------

#define DK 256          // feature dim
#define BM 128          // rows per block (8 waves x 16)
#define BN 64           // cols per block
#define PSTRIDE 80      // float2 stride per K-pair row: 160 floats -> bank shift 32
#define GEMM_LDS_BYTES ((DK / 2) * PSTRIDE * 8)   // 81,920 B (dynamic LDS)

typedef __attribute__((ext_vector_type(2))) float v2f;
typedef __attribute__((ext_vector_type(4))) float v4f;
typedef __attribute__((ext_vector_type(8))) float v8f;

// ---------------------------------------------------------------------------
// Y[M x 256] = (A[M x 256] * pre(row)) @ W[256 x 256] + bias, * post(row)
//   PRE : A row scaled by 1/(cnt[row] + 1e-8)  -> fuses scatter-mean
//   POST: output row scaled by w[row]          -> fuses hyperedge weight
// One wave computes a 16x64 strip via 4x V_WMMA_F32_16X16X4_F32 per K-step.
// ---------------------------------------------------------------------------
template<bool PRE, bool POST>
__global__ __launch_bounds__(256, 2)
void hg_gemm_wmma(const float* __restrict__ A, const float* __restrict__ W,
                  const float* __restrict__ bias, const float* __restrict__ cnt_pre,
                  const float* __restrict__ scale_post, float* __restrict__ Y, int M)
{
  extern __shared__ float wlds[];   // [DK/2 pairs][PSTRIDE float2]
  const int tid  = threadIdx.x;
  const int wave = tid >> 5;
  const int lane = tid & 31;
  const int hi   = lane >> 4;       // half-wave: K offset +2 (next K pair)
  const int l15  = lane & 15;
  const int nbase = blockIdx.y * BN;
  const int mbase = blockIdx.x * BM + wave * 16;

  // Stage W[:, nbase : nbase+64] into LDS, K-pair-major:
  // float2 at [p][n] = { W[2p][n], W[2p+1][n] }. Two ds_store_b128 per item.
#pragma unroll
  for (int it = 0; it < ((DK / 2) * (BN / 4)) / 256; ++it) {   // 8 iterations
    const int e  = it * 256 + tid;
    const int p  = e >> 4;                    // K pair index (0..127)
    const int n4 = e & 15;                    // group of 4 columns
    const v4f r0 = *(const v4f*)(W + (size_t)(2 * p)     * DK + nbase + n4 * 4);
    const v4f r1 = *(const v4f*)(W + (size_t)(2 * p + 1) * DK + nbase + n4 * 4);
    float* dst = &wlds[p * (PSTRIDE * 2) + n4 * 8];
    *(v4f*)(dst)     = (v4f){ r0.x, r1.x, r0.y, r1.y };
    *(v4f*)(dst + 4) = (v4f){ r0.z, r1.z, r0.w, r1.w };
  }

  // A fragment addressing (16x4 fp32 layout: lanes 0-15 K={0,1}, lanes 16-31 K={2,3}).
  // Out-of-range rows are clamped to row 0: they only feed out-of-range C rows,
  // which are never stored, so no zeroing is needed.
  const int  arow_i = mbase + l15;
  const int  arow_c = arow_i < M ? arow_i : 0;
  const float* __restrict__ Ar = A + (size_t)arow_c * DK + hi * 2;

  // Warm the back half of this wave's A rows while we wait at the barrier
  // (single global_prefetch_b8, kept OUT of the hot loop -- no per-iteration
  // branch/SALU overhead).
  __builtin_prefetch(Ar + DK / 2, 0, 3);

  float pre = 1.0f;
  if (PRE) pre = 1.0f / (cnt_pre[arow_c] + 1e-8f);

  __syncthreads();

  v8f c0 = {0,0,0,0,0,0,0,0};
  v8f c1 = c0, c2 = c0, c3 = c0;

  for (int k = 0; k < DK; k += 4) {
    v2f a = *(const v2f*)(Ar + k);
    if (PRE) { a.x *= pre; a.y *= pre; }

    // B fragments: single aligned b64 per tile; pair p for lanes 0-15, p+1 for 16-31.
    const float* bp = &wlds[((k >> 1) + hi) * (PSTRIDE * 2) + l15 * 2];
    v2f b0 = *(const v2f*)(bp);        // n0 = 0
    v2f b1 = *(const v2f*)(bp + 32);   // n0 = 16
    v2f b2 = *(const v2f*)(bp + 64);   // n0 = 32
    v2f b3 = *(const v2f*)(bp + 96);   // n0 = 48

    c0 = __builtin_amdgcn_wmma_f32_16x16x4_f32(false, a, false, b0, (short)0, c0, false, false);
    c1 = __builtin_amdgcn_wmma_f32_16x16x4_f32(false, a, false, b1, (short)0, c1, false, false);
    c2 = __builtin_amdgcn_wmma_f32_16x16x4_f32(false, a, false, b2, (short)0, c2, false, false);
    c3 = __builtin_amdgcn_wmma_f32_16x16x4_f32(false, a, false, b3, (short)0, c3, false, false);
  }

  // Epilogue: C layout -> VGPR r holds (M = mbase + hi*8 + r, N = nbase + n0 + l15)
  const float bn0 = bias[nbase + 0  + l15];
  const float bn1 = bias[nbase + 16 + l15];
  const float bn2 = bias[nbase + 32 + l15];
  const float bn3 = bias[nbase + 48 + l15];
#pragma unroll
  for (int r = 0; r < 8; ++r) {
    const int orow = mbase + hi * 8 + r;
    if (orow < M) {
      float ps = 1.0f;
      if (POST) ps = scale_post[orow];
      float* yr = Y + (size_t)orow * DK + nbase + l15;
      yr[0]  = (c0[r] + bn0) * ps;
      yr[16] = (c1[r] + bn1) * ps;
      yr[32] = (c2[r] + bn2) * ps;
      yr[48] = (c3[r] + bn3) * ps;
    }
  }
}

// ---------------------------------------------------------------------------
// Zero a float buffer (n must be a multiple of 4 -- true for all our buffers).
// ---------------------------------------------------------------------------
__global__ __launch_bounds__(256)
void hg_zero(float* __restrict__ p, size_t n)
{
  const size_t stride = (size_t)gridDim.x * 256 * 4;
  for (size_t i = ((size_t)blockIdx.x * 256 + threadIdx.x) * 4; i < n; i += stride)
    *(v4f*)(p + i) = (v4f){0.f, 0.f, 0.f, 0.f};
}

// ---------------------------------------------------------------------------
// For each connection c: dst[sidx[c]] += src[gidx[c]] (256-wide rows, f32
// global atomics resolved in L2); cnt[sidx[c]] += 1.
// ---------------------------------------------------------------------------
__global__ __launch_bounds__(256)
void hg_scatter(const float* __restrict__ src, const int* __restrict__ gidx,
                const int* __restrict__ sidx, float* __restrict__ dst,
                float* __restrict__ cnt, int nconn)
{
  for (int c = blockIdx.x; c < nconn; c += gridDim.x) {
    const int g = gidx[c];
    const int s = sidx[c];
    const float v = src[(size_t)g * DK + threadIdx.x];   // coalesced 1 KB row read
    atomicAdd(dst + (size_t)s * DK + threadIdx.x, v);
    if (threadIdx.x == 0) atomicAdd(cnt + s, 1.0f);
  }
}

__device__ __forceinline__ float lrelu(float v) { return v > 0.0f ? v : 0.2f * v; }

// ---------------------------------------------------------------------------
// Per node row: agg = (cnt>0) ? nsum/cnt : 0 ; h = agg + xt ;
// LayerNorm(h)*g + b ; LeakyReLU(0.2) ; (+ res for layers > 0).
// One wave32 per row; lane owns 8 contiguous elements; shfl_xor reductions.
// ---------------------------------------------------------------------------
__global__ __launch_bounds__(256)
void hg_finalize(const float* __restrict__ nsum, const float* __restrict__ ncnt,
                 const float* __restrict__ xt, const float* __restrict__ g,
                 const float* __restrict__ b, const float* __restrict__ res,
                 float* __restrict__ out, int M)
{
  const int row  = blockIdx.x * 8 + (threadIdx.x >> 5);
  const int lane = threadIdx.x & 31;
  if (row >= M) return;

  const size_t base = (size_t)row * DK + lane * 8;
  const float cnt = ncnt[row];
  const float inv = cnt > 0.0f ? 1.0f / cnt : 0.0f;   // cnt is integral >= 1 when > 0

  v4f h0 = *(const v4f*)(nsum + base);
  v4f h1 = *(const v4f*)(nsum + base + 4);
  h0 = h0 * inv + *(const v4f*)(xt + base);
  h1 = h1 * inv + *(const v4f*)(xt + base + 4);

  float s = h0.x + h0.y + h0.z + h0.w + h1.x + h1.y + h1.z + h1.w;
  float q = h0.x*h0.x + h0.y*h0.y + h0.z*h0.z + h0.w*h0.w
          + h1.x*h1.x + h1.y*h1.y + h1.z*h1.z + h1.w*h1.w;
#pragma unroll
  for (int off = 16; off > 0; off >>= 1) {
    s += __shfl_xor(s, off, 32);
    q += __shfl_xor(q, off, 32);
  }
  const float mu  = s * (1.0f / 256.0f);
  const float var = q * (1.0f / 256.0f) - mu * mu;     // biased var (matches jnp.var)
  const float rs  = rsqrtf(var + 1e-5f);

  const v4f g0 = *(const v4f*)(g + lane * 8);
  const v4f g1 = *(const v4f*)(g + lane * 8 + 4);
  const v4f b0 = *(const v4f*)(b + lane * 8);
  const v4f b1 = *(const v4f*)(b + lane * 8 + 4);

  v4f y0 = (h0 - mu) * rs * g0 + b0;
  v4f y1 = (h1 - mu) * rs * g1 + b1;
  y0.x = lrelu(y0.x); y0.y = lrelu(y0.y); y0.z = lrelu(y0.z); y0.w = lrelu(y0.w);
  y1.x = lrelu(y1.x); y1.y = lrelu(y1.y); y1.z = lrelu(y1.z); y1.w = lrelu(y1.w);

  if (res) {
    y0 += *(const v4f*)(res + base);
    y1 += *(const v4f*)(res + base + 4);
  }
  *(v4f*)(out + base)     = y0;
  *(v4f*)(out + base + 4) = y1;
}

// ---------------------------------------------------------------------------
// Host-side orchestration (graph-capture safe: stream-only launches, ws scratch)
// ---------------------------------------------------------------------------
extern "C" void kernel_launch(void* const* d_in, const int* in_sizes, int n_in,
                              void* d_out, int out_size, void* d_ws, size_t ws_size,
                              hipStream_t stream)
{
  const float* x       = (const float*)d_in[0];
  const float* nodeW   = (const float*)d_in[1];
  const float* nodeB   = (const float*)d_in[2];
  const float* heW     = (const float*)d_in[3];
  const float* heB     = (const float*)d_in[4];
  const float* lnG     = (const float*)d_in[5];
  const float* lnB     = (const float*)d_in[6];
  const float* heWt    = (const float*)d_in[7];
  const int*   nodeIdx = (const int*)d_in[8];
  const int*   heIdx   = (const int*)d_in[9];

  const int Nn  = in_sizes[0] / DK;   // 100000 nodes
  const int Nhe = in_sizes[7];        // 25000 hyperedges
  const int Nc  = in_sizes[8];        // 400000 connections
  const int L   = in_sizes[2] / DK;   // 2 layers

  // Workspace carve-out (256 B aligned): ~359 MB total.
  char* wsp = (char*)d_ws;
  auto carve = [&](size_t bytes) -> float* {
    float* p = (float*)wsp;
    wsp += (bytes + 255) & ~(size_t)255;
    return p;
  };
  float* xt    = carve((size_t)Nn  * DK * 4);   // node_transform output
  float* nsum  = carve((size_t)Nn  * DK * 4);   // he -> node scatter target
  float* xping = carve((size_t)Nn  * DK * 4);   // inter-layer activations
  float* hesum = carve((size_t)Nhe * DK * 4);   // node -> he scatter target
  float* he2   = carve((size_t)Nhe * DK * 4);   // hyperedge_transform output
  float* hecnt = carve((size_t)Nhe * 4);
  float* ncnt  = carve((size_t)Nn  * 4);

  float* outp = (float*)d_out;

  const dim3 gemmGridN((Nn  + BM - 1) / BM, DK / BN);   // 782 x 4
  const dim3 gemmGridH((Nhe + BM - 1) / BM, DK / BN);   // 196 x 4

  const float* xcur = x;
  for (int l = 0; l < L; ++l) {
    const size_t wOff = (size_t)l * DK * DK;
    const size_t vOff = (size_t)l * DK;

    // xt = x @ Wn + bn
    hg_gemm_wmma<false, false><<<gemmGridN, 256, GEMM_LDS_BYTES, stream>>>(
        xcur, nodeW + wOff, nodeB + vOff, nullptr, nullptr, xt, Nn);

    // nodes -> hyperedges (scatter-sum + counts)
    hg_zero<<<1024, 256, 0, stream>>>(hesum, (size_t)Nhe * DK);
    hg_zero<<<64,   256, 0, stream>>>(hecnt, (size_t)Nhe);
    hg_scatter<<<4096, 256, 0, stream>>>(xt, nodeIdx, heIdx, hesum, hecnt, Nc);

    // he2 = ((hesum / (cnt+eps)) @ We + be) * w   (mean + weight fused into GEMM)
    hg_gemm_wmma<true, true><<<gemmGridH, 256, GEMM_LDS_BYTES, stream>>>(
        hesum, heW + wOff, heB + vOff, hecnt, heWt, he2, Nhe);

    // hyperedges -> nodes (scatter-sum + counts)
    hg_zero<<<2048, 256, 0, stream>>>(nsum, (size_t)Nn * DK);
    hg_zero<<<128,  256, 0, stream>>>(ncnt, (size_t)Nn);
    hg_scatter<<<4096, 256, 0, stream>>>(he2, heIdx, nodeIdx, nsum, ncnt, Nc);

    // mean + residual(xt) + LayerNorm + LeakyReLU (+ layer residual for l>0)
    float* dst = (l == L - 1) ? outp : xping;
    hg_finalize<<<(Nn + 7) / 8, 256, 0, stream>>>(
        nsum, ncnt, xt, lnG + vOff, lnB + vOff,
        (l > 0) ? xcur : nullptr, dst, Nn);
    xcur = dst;
  }
}